// ShallowGNNAutoEncoder_88313117541119
// MI455X (gfx1250) — compile-verified
//
#include <hip/hip_runtime.h>

// =====================================================================
// GCN autoencoder for MI455X (gfx1250, wave32).
//   enc: h = x@W_enc ; z = D^-1/2 A_hat D^-1/2 h + b_enc   (C=64)
//   dec: h2 = z@W_dec; out = D^-1/2 A_hat D^-1/2 h2 + b_dec (C=128)
// GEMMs (negligible FLOPs -> keep full fp32) use V_WMMA_F32_16X16X4_F32,
// one 16 x (NT*16) output strip per wave: A fragment loaded once per
// k-step (global b64) and reused across NT WMMAs; B held transposed in
// padded LDS so each fragment is one ds_load_b64, bank-conflict-free.
// Scatter phase is L2-bandwidth bound (working set < 192MB L2) and uses
// native global_atomic_add_f32.
// Assumes N_NODES % 16 == 0 (50000 = 3125*16) and edge_index as int32.
// =====================================================================

typedef __attribute__((ext_vector_type(2))) float v2f;
typedef __attribute__((ext_vector_type(8))) float v8f;

// ---------------- degree / normalization ----------------
__global__ void deg_init_k(float* __restrict__ deg, int n) {
  int i = blockIdx.x * blockDim.x + threadIdx.x;
  if (i < n) deg[i] = 1.0f;  // self-loop contribution
}

__global__ void deg_count_k(const int* __restrict__ dst, float* __restrict__ deg, int e) {
  int i = blockIdx.x * blockDim.x + threadIdx.x;
  if (i < e) unsafeAtomicAdd(&deg[dst[i]], 1.0f);
}

__global__ void deg_to_dinv_k(float* __restrict__ d, int n) {
  int i = blockIdx.x * blockDim.x + threadIdx.x;
  if (i < n) d[i] = rsqrtf(d[i]);  // deg >= 1 always (self-loops)
}

// ---------------- WMMA f32 GEMM: C[M, NT*16] = A[M,K] @ B[K, NT*16] -------
// A fragment (16x4): lane%16 = M row; lanes<16 -> K=k,k+1 ; lanes>=16 -> k+2,k+3
// B fragment (4x16): lane%16 = N col; same K split. Transposed-in-LDS makes
// (b.x,b.y) an adjacent pair -> single ds_load_b64.
// C/D: acc[rr] -> row strip*16 + (lane<16 ? rr : 8+rr), col t*16 + lane%16.
template <int K, int NT>
__global__ void gemm_f32_wmma_k(const float* __restrict__ A, const float* __restrict__ B,
                                float* __restrict__ C, int Mrows) {
  constexpr int NC  = NT * 16;   // output columns
  constexpr int LDB = K + 2;     // padded stride: banks advance by 2 per column,
                                 // (K+2)*4 bytes stays 8B-aligned for b64 loads
  __shared__ __align__(16) float sBt[NC * LDB];

  const int tid = threadIdx.x;
  // cooperative transpose: B row-major [K x NC] -> sBt[n*LDB + k]
  for (int i = tid; i < K * NC; i += blockDim.x) {
    int kidx = i / NC;
    int n = i - kidx * NC;
    sBt[n * LDB + kidx] = B[i];
  }
  __syncthreads();

  const int strips = Mrows >> 4;
  const int strip = blockIdx.x * (blockDim.x >> 5) + (tid >> 5);
  if (strip >= strips) return;   // wave-uniform; after barrier

  const int lane = tid & 31;
  const int r    = lane & 15;
  const int kk   = (lane >> 4) << 1;   // 0 or 2

  const float* ap = A + (long long)(strip * 16 + r) * K + kk;

  v8f acc[NT];
#pragma unroll
  for (int t = 0; t < NT; ++t) acc[t] = (v8f){0.f,0.f,0.f,0.f,0.f,0.f,0.f,0.f};

  for (int k = 0; k < K; k += 4) {
    v2f a = *(const v2f*)(ap + k);                     // global_load_b64
#pragma unroll
    for (int t = 0; t < NT; ++t) {
      v2f b = *(const v2f*)&sBt[(t * 16 + r) * LDB + k + kk];  // ds_load_b64
      acc[t] = __builtin_amdgcn_wmma_f32_16x16x4_f32(
          /*neg_a=*/false, a, /*neg_b=*/false, b,
          /*c_mod=*/(short)0, acc[t], /*reuse_a=*/false, /*reuse_b=*/false);
    }
  }

  const int mBase = strip * 16 + ((lane >> 4) << 3);
#pragma unroll
  for (int t = 0; t < NT; ++t) {
    float* cp = C + (long long)mBase * NC + t * 16 + r;
#pragma unroll
    for (int rr = 0; rr < 8; ++rr) cp[(long long)rr * NC] = acc[t][rr];
  }
}

// ---------------- out = dinv^2 * h + bias (self-loop term, full init) -----
template <int C>
__global__ void init_out_k(const float* __restrict__ h, const float* __restrict__ dinv,
                           const float* __restrict__ bias, float* __restrict__ out, int n) {
  int idx = blockIdx.x * blockDim.x + threadIdx.x;
  if (idx >= n * C) return;
  int i = idx / C;           // power of two -> shift
  int c = idx & (C - 1);
  float di = dinv[i];
  out[idx] = di * di * h[idx] + bias[c];
}

// ---------------- edge scatter: out[dst] += dinv[s]*dinv[d]*h[src] --------
template <int C>
__global__ void edge_scatter_k(const int* __restrict__ src, const int* __restrict__ dst,
                               const float* __restrict__ dinv, const float* __restrict__ h,
                               float* __restrict__ out, int E) {
  constexpr int C4 = C / 4;
  int idx = blockIdx.x * blockDim.x + threadIdx.x;
  if (idx >= E * C4) return;
  int e  = idx / C4;         // power of two -> shift
  int c4 = idx & (C4 - 1);
  int s = src[e], d = dst[e];
  float norm = dinv[s] * dinv[d];
  const float4 hv = reinterpret_cast<const float4*>(h + (long long)s * C)[c4];
  float* op = out + (long long)d * C + (c4 << 2);
  unsafeAtomicAdd(op + 0, norm * hv.x);
  unsafeAtomicAdd(op + 1, norm * hv.y);
  unsafeAtomicAdd(op + 2, norm * hv.z);
  unsafeAtomicAdd(op + 3, norm * hv.w);
}

// =====================================================================
extern "C" void kernel_launch(void* const* d_in, const int* in_sizes, int n_in,
                              void* d_out, int out_size, void* d_ws, size_t ws_size,
                              hipStream_t stream) {
  const float* x     = (const float*)d_in[0];
  const int*   edges = (const int*)d_in[1];   // [2, E], int32 per harness convention
  const float* W_enc = (const float*)d_in[2]; // [128, 64]
  const float* b_enc = (const float*)d_in[3]; // [64]
  const float* W_dec = (const float*)d_in[4]; // [64, 128]
  const float* b_dec = (const float*)d_in[5]; // [128]
  float* outF = (float*)d_out;

  const int IN_CH = 128, BOT = 64;
  const int Nn = in_sizes[0] / IN_CH;   // 50000
  const int E  = in_sizes[1] / 2;       // 800000
  const int* src = edges;
  const int* dst = edges + E;

  // workspace layout (16B-aligned)
  char* ws = (char*)d_ws;
  size_t off = 0;
  float* dinv = (float*)(ws + off); off += ((size_t)Nn * 4 + 255) & ~(size_t)255;
  float* h    = (float*)(ws + off); off += (size_t)Nn * BOT * 4;     // [N,64]
  float* z    = (float*)(ws + off); off += (size_t)Nn * BOT * 4;     // [N,64]
  float* h2   = (float*)(ws + off); off += (size_t)Nn * IN_CH * 4;   // [N,128]

  const int T = 256;  // 8 waves per block

  // ---- symmetric normalization coefficients
  deg_init_k<<<(Nn + T - 1) / T, T, 0, stream>>>(dinv, Nn);
  deg_count_k<<<(E + T - 1) / T, T, 0, stream>>>(dst, dinv, E);
  deg_to_dinv_k<<<(Nn + T - 1) / T, T, 0, stream>>>(dinv, Nn);

  const int strips = Nn / 16;                 // 3125
  const int gemmBlocks = (strips + 7) / 8;    // 8 waves/block, 1 strip/wave

  // ---- encoder: h = x @ W_enc   (K=128, NT=4 -> 64 cols)
  gemm_f32_wmma_k<128, 4><<<gemmBlocks, T, 0, stream>>>(x, W_enc, h, Nn);
  init_out_k<64><<<(Nn * 64 + T - 1) / T, T, 0, stream>>>(h, dinv, b_enc, z, Nn);
  edge_scatter_k<64><<<(E * 16 + T - 1) / T, T, 0, stream>>>(src, dst, dinv, h, z, E);

  // ---- decoder: h2 = z @ W_dec  (K=64, NT=8 -> 128 cols)
  gemm_f32_wmma_k<64, 8><<<gemmBlocks, T, 0, stream>>>(z, W_dec, h2, Nn);
  init_out_k<128><<<(Nn * 128 + T - 1) / T, T, 0, stream>>>(h2, dinv, b_dec, outF, Nn);
  edge_scatter_k<128><<<(E * 32 + T - 1) / T, T, 0, stream>>>(src, dst, dinv, h2, outF, E);
}